// GraphNet_43499428774654
// MI455X (gfx1250) — compile-verified
//
#include <hip/hip_runtime.h>
#include <hip/hip_bf16.h>

// ---------------------------------------------------------------------------
// GraphNet forward on gfx1250 (MI455X), fp32 WMMA (V_WMMA_F32_16X16X4_F32).
// B=64, N=128, H=32, HE=64, OE=32, E_IN=66, 4 MP steps.
// Edge MLP = 1,048,576 rows -> recompute strategy across 3 passes so
// workspace stays ~4 MB (BatchNorm needs global stats between layers).
// ---------------------------------------------------------------------------

typedef float v2f __attribute__((ext_vector_type(2)));
typedef float v8f __attribute__((ext_vector_type(8)));

__device__ __forceinline__ float lrelu_f(float v) { return v > 0.f ? v : 0.1f * v; }

__device__ __forceinline__ v8f wmma4(v2f a, v2f b, v8f c) {
  // 8-arg form: (neg_a, A, neg_b, B, c_mod, C, reuse_a, reuse_b)
  return __builtin_amdgcn_wmma_f32_16x16x4_f32(false, a, false, b, (short)0, c, false, false);
}

// ---------------------------------------------------------------------------
// Edge pass kernel. One block = one (b, i) pair; 128 edge rows j=0..127.
// 256 threads = 8 waves; wave w owns rows [16w, 16w+16).
// PASS 1: GEMM1 + lrelu + stats(64)            -> sum1/sq1
// PASS 2: GEMM1 + BN1 + GEMM2 + lrelu + stats  -> sum2/sq2
// PASS 3: GEMM1 + BN1 + GEMM2 + BN2 + sum_j    -> agg[b,i,:]
// ---------------------------------------------------------------------------
template <int PASS>
__global__ __launch_bounds__(256) void edge_pass_kernel(
    const float* __restrict__ xcur,      // [B*N*32]
    const float* __restrict__ W1,        // [66*64]
    const float* __restrict__ b1,        // [64]
    const float* __restrict__ W2,        // [64*32]
    const float* __restrict__ b2,        // [32]
    const float* __restrict__ scale1, const float* __restrict__ shift1,  // [64]
    const float* __restrict__ scale2, const float* __restrict__ shift2,  // [32]
    float* __restrict__ sum1, float* __restrict__ sq1,   // [64]
    float* __restrict__ sum2, float* __restrict__ sq2,   // [32]
    float* __restrict__ agg)             // [B*N*32]
{
  __shared__ float s_union[8832];  // phaseA: xs_j[128*34] dist[128] int[128] W1[66*64]
                                   // phaseB: zbuf[128*66]
  __shared__ float s_xi[32];
  __shared__ float s_w2[2048];
  __shared__ float s_red[128];

  const int tid  = threadIdx.x;
  const int lane = tid & 31;
  const int wv   = tid >> 5;       // 0..7
  const int h    = lane >> 4;      // 0/1
  const int m    = lane & 15;
  const int row0 = wv * 16;

  const int e = blockIdx.x;        // b*128 + i
  const int b = e >> 7;

  float* xs_j   = s_union;               // stride 34
  float* s_dist = s_union + 4352;
  float* s_int  = s_union + 4480;
  float* s_w1   = s_union + 4608;        // 66*64 = 4224 floats
  float* zbuf   = s_union;               // stride 66 (phase B)

  if (tid < 128) s_red[tid] = 0.f;
  if (tid < 32)  s_xi[tid] = xcur[e * 32 + tid];

  const float* xb = xcur + b * 128 * 32;
  for (int idx = tid; idx < 4096; idx += 256) {
    int r = idx >> 5, c = idx & 31;
    xs_j[r * 34 + c] = xb[r * 32 + c];
  }
  for (int idx = tid; idx < 4224; idx += 256) s_w1[idx] = W1[idx];
  if (PASS >= 2)
    for (int idx = tid; idx < 2048; idx += 256) s_w2[idx] = W2[idx];

  __syncthreads();

  // per-row distance + intensity-diff features
  if (tid < 128) {
    float acc = 0.f;
    #pragma unroll
    for (int k = 0; k < 31; ++k) {
      float d = xs_j[tid * 34 + k] - s_xi[k] + 1e-12f;
      acc += d * d;
    }
    s_dist[tid] = sqrtf(acc);
    s_int[tid]  = 1.0f - (xs_j[tid * 34 + 31] - s_xi[31]);
  }
  __syncthreads();

  // ---- GEMM1: [16 x 66] @ [66 x 64], K in chunks of 4 -----------------
  float sc1[4], sh1[4];
  v8f acc1[4];
  #pragma unroll
  for (int c = 0; c < 4; ++c) {
    float bv = b1[16 * c + m];
    #pragma unroll
    for (int g = 0; g < 8; ++g) acc1[c][g] = bv;
    if (PASS >= 2) { sc1[c] = scale1[16 * c + m]; sh1[c] = shift1[16 * c + m]; }
  }

  #pragma unroll 4
  for (int kc = 0; kc < 16; ++kc) {
    int kk = 4 * kc + 2 * h;
    v2f a;
    if (kc < 8) { a.x = s_xi[kk]; a.y = s_xi[kk + 1]; }               // sender feats
    else {                                                            // receiver feats
      const float* p = &xs_j[(row0 + m) * 34 + (kk - 32)];
      a.x = p[0]; a.y = p[1];
    }
    #pragma unroll
    for (int c = 0; c < 4; ++c) {
      v2f bf;
      bf.x = s_w1[(kk + 0) * 64 + 16 * c + m];
      bf.y = s_w1[(kk + 1) * 64 + 16 * c + m];
      acc1[c] = wmma4(a, bf, acc1[c]);
    }
  }
  { // tail chunk: K = 64..67 -> [dist, intdiff, 0, 0]
    v2f a;
    a.x = (h == 0) ? s_dist[row0 + m] : 0.f;
    a.y = (h == 0) ? s_int[row0 + m]  : 0.f;
    #pragma unroll
    for (int c = 0; c < 4; ++c) {
      v2f bf;
      bf.x = (h == 0) ? s_w1[64 * 64 + 16 * c + m] : 0.f;
      bf.y = (h == 0) ? s_w1[65 * 64 + 16 * c + m] : 0.f;
      acc1[c] = wmma4(a, bf, acc1[c]);
    }
  }

  // lrelu
  #pragma unroll
  for (int c = 0; c < 4; ++c)
    #pragma unroll
    for (int g = 0; g < 8; ++g) acc1[c][g] = lrelu_f(acc1[c][g]);

  if (PASS == 1) {  // stats of layer-1 activations
    #pragma unroll
    for (int c = 0; c < 4; ++c) {
      float ps = 0.f, pq = 0.f;
      #pragma unroll
      for (int g = 0; g < 8; ++g) { float v = acc1[c][g]; ps += v; pq += v * v; }
      atomicAdd(&s_red[16 * c + m], ps);
      atomicAdd(&s_red[64 + 16 * c + m], pq);
    }
    __syncthreads();
    if (tid < 64) { atomicAdd(&sum1[tid], s_red[tid]); atomicAdd(&sq1[tid], s_red[64 + tid]); }
    return;
  }

  // BN1 apply
  #pragma unroll
  for (int c = 0; c < 4; ++c)
    #pragma unroll
    for (int g = 0; g < 8; ++g) acc1[c][g] = acc1[c][g] * sc1[c] + sh1[c];

  __syncthreads();  // everyone done reading xs_j/W1 -> reuse region for zbuf
  #pragma unroll
  for (int c = 0; c < 4; ++c)
    #pragma unroll
    for (int g = 0; g < 8; ++g)
      zbuf[(row0 + g + 8 * h) * 66 + 16 * c + m] = acc1[c][g];
  __syncthreads();

  // ---- GEMM2: [16 x 64] @ [64 x 32] -----------------------------------
  float sc2[2], sh2[2];
  v8f acc2[2];
  #pragma unroll
  for (int c = 0; c < 2; ++c) {
    float bv = b2[16 * c + m];
    #pragma unroll
    for (int g = 0; g < 8; ++g) acc2[c][g] = bv;
    if (PASS == 3) { sc2[c] = scale2[16 * c + m]; sh2[c] = shift2[16 * c + m]; }
  }
  #pragma unroll 4
  for (int kc = 0; kc < 16; ++kc) {
    int kk = 4 * kc + 2 * h;
    v2f a;
    const float* p = &zbuf[(row0 + m) * 66 + kk];
    a.x = p[0]; a.y = p[1];
    #pragma unroll
    for (int c = 0; c < 2; ++c) {
      v2f bf;
      bf.x = s_w2[(kk + 0) * 32 + 16 * c + m];
      bf.y = s_w2[(kk + 1) * 32 + 16 * c + m];
      acc2[c] = wmma4(a, bf, acc2[c]);
    }
  }
  #pragma unroll
  for (int c = 0; c < 2; ++c)
    #pragma unroll
    for (int g = 0; g < 8; ++g) acc2[c][g] = lrelu_f(acc2[c][g]);

  if (PASS == 2) {  // stats of layer-2 activations
    #pragma unroll
    for (int c = 0; c < 2; ++c) {
      float ps = 0.f, pq = 0.f;
      #pragma unroll
      for (int g = 0; g < 8; ++g) { float v = acc2[c][g]; ps += v; pq += v * v; }
      atomicAdd(&s_red[16 * c + m], ps);
      atomicAdd(&s_red[64 + 16 * c + m], pq);
    }
    __syncthreads();
    if (tid < 32) { atomicAdd(&sum2[tid], s_red[tid]); atomicAdd(&sq2[tid], s_red[64 + tid]); }
    return;
  }

  // PASS 3: BN2 then sum over all 128 receivers j
  #pragma unroll
  for (int c = 0; c < 2; ++c) {
    float ps = 0.f;
    #pragma unroll
    for (int g = 0; g < 8; ++g) ps += acc2[c][g] * sc2[c] + sh2[c];
    atomicAdd(&s_red[16 * c + m], ps);
  }
  __syncthreads();
  if (tid < 32) agg[e * 32 + tid] = s_red[tid];
}

// ---------------------------------------------------------------------------
// Node MLP layer: Y = act(X @ W + b), X is [8192 x K] (K=64 concat or 32),
// W is [K x 32]. osum!=null: lrelu + accumulate BN stats; else tanh (final).
// One block = 128 rows, 8 waves of 16 rows.
// ---------------------------------------------------------------------------
__global__ __launch_bounds__(256) void node_layer_kernel(
    const float* __restrict__ X1, const float* __restrict__ X2,  // X2!=null => concat(X1,X2)
    const float* __restrict__ scale_in, const float* __restrict__ shift_in,  // BN applied on load
    const float* __restrict__ W, const float* __restrict__ bias, int K,
    float* __restrict__ Y,
    float* __restrict__ osum, float* __restrict__ osq)
{
  __shared__ float xt[128 * 66];
  __shared__ float wt[2048];
  __shared__ float s_red[64];

  const int tid  = threadIdx.x;
  const int lane = tid & 31;
  const int wv   = tid >> 5;
  const int h    = lane >> 4;
  const int m    = lane & 15;
  const int row0 = wv * 16;
  const int gbase = blockIdx.x * 128;
  const int stride = K + 2;

  if (tid < 64) s_red[tid] = 0.f;

  for (int idx = tid; idx < 128 * K; idx += 256) {
    int r = idx / K, c = idx - r * K;
    float v;
    if (X2) v = (c < 32) ? X1[(gbase + r) * 32 + c] : X2[(gbase + r) * 32 + (c - 32)];
    else {
      v = X1[(gbase + r) * K + c];
      if (scale_in) v = v * scale_in[c] + shift_in[c];
    }
    xt[r * stride + c] = v;
  }
  for (int idx = tid; idx < K * 32; idx += 256) wt[idx] = W[idx];
  __syncthreads();

  v8f acc[2];
  #pragma unroll
  for (int c = 0; c < 2; ++c) {
    float bv = bias[16 * c + m];
    #pragma unroll
    for (int g = 0; g < 8; ++g) acc[c][g] = bv;
  }
  const int KC = K >> 2;
  for (int kc = 0; kc < KC; ++kc) {
    int kk = 4 * kc + 2 * h;
    v2f a;
    const float* p = &xt[(row0 + m) * stride + kk];
    a.x = p[0]; a.y = p[1];
    #pragma unroll
    for (int c = 0; c < 2; ++c) {
      v2f bf;
      bf.x = wt[(kk + 0) * 32 + 16 * c + m];
      bf.y = wt[(kk + 1) * 32 + 16 * c + m];
      acc[c] = wmma4(a, bf, acc[c]);
    }
  }

  if (osum) {
    #pragma unroll
    for (int c = 0; c < 2; ++c) {
      float ps = 0.f, pq = 0.f;
      #pragma unroll
      for (int g = 0; g < 8; ++g) {
        float v = lrelu_f(acc[c][g]);
        acc[c][g] = v; ps += v; pq += v * v;
      }
      atomicAdd(&s_red[16 * c + m], ps);
      atomicAdd(&s_red[32 + 16 * c + m], pq);
    }
    #pragma unroll
    for (int c = 0; c < 2; ++c)
      #pragma unroll
      for (int g = 0; g < 8; ++g)
        Y[(gbase + row0 + g + 8 * h) * 32 + 16 * c + m] = acc[c][g];
    __syncthreads();
    if (tid < 32) { atomicAdd(&osum[tid], s_red[tid]); atomicAdd(&osq[tid], s_red[32 + tid]); }
  } else {
    #pragma unroll
    for (int c = 0; c < 2; ++c)
      #pragma unroll
      for (int g = 0; g < 8; ++g)
        Y[(gbase + row0 + g + 8 * h) * 32 + 16 * c + m] = tanhf(acc[c][g]);
  }
}

// sum/sumsq -> BN (scale, shift) with biased variance, eps 1e-5
__global__ void finalize_stats_kernel(const float* __restrict__ sum, const float* __restrict__ sq,
                                      const float* __restrict__ gamma, const float* __restrict__ beta,
                                      float* __restrict__ scale, float* __restrict__ shift,
                                      int n, float cnt)
{
  int t = threadIdx.x;
  if (t < n) {
    float mu  = sum[t] / cnt;
    float var = sq[t] / cnt - mu * mu;
    float inv = rsqrtf(var + 1e-5f);
    float sc  = gamma[t] * inv;
    scale[t] = sc;
    shift[t] = beta[t] - mu * sc;
  }
}

// pad [B*N,3] -> [B*N,32]
__global__ void pad_init_kernel(const float* __restrict__ xin, float* __restrict__ xcur)
{
  int idx = blockIdx.x * 256 + threadIdx.x;  // 262144 total
  int r = idx >> 5, c = idx & 31;
  xcur[idx] = (c < 3) ? xin[r * 3 + c] : 0.f;
}

extern "C" void kernel_launch(void* const* d_in, const int* in_sizes, int n_in,
                              void* d_out, int out_size, void* d_ws, size_t ws_size,
                              hipStream_t stream)
{
  const float* x    = (const float*)d_in[0];
  const float* Wah  = (const float*)d_in[1];
  const float* bah  = (const float*)d_in[2];
  const float* Wa   = (const float*)d_in[3];
  const float* ba   = (const float*)d_in[4];
  const float* g_eh = (const float*)d_in[5];
  const float* b_eh = (const float*)d_in[6];
  const float* g_e  = (const float*)d_in[7];
  const float* b_e  = (const float*)d_in[8];
  const float* Wn0  = (const float*)d_in[9];
  const float* bn0  = (const float*)d_in[10];
  const float* Wn   = (const float*)d_in[11];
  const float* bnn  = (const float*)d_in[12];
  const float* g_n  = (const float*)d_in[13];
  const float* b_n  = (const float*)d_in[14];
  const float* Wu   = (const float*)d_in[15];
  const float* bu   = (const float*)d_in[16];

  float* ws   = (float*)d_ws;
  float* xcur = ws;                 // 262144
  float* agg  = ws + 262144;        // 262144
  float* h1   = ws + 524288;        // 262144
  float* h2   = ws + 786432;        // 262144
  float* st   = ws + 1048576;       // 640 floats of stats
  float *sum1 = st,       *sq1 = st + 64,  *scale1 = st + 128, *shift1 = st + 192;
  float *sum2 = st + 256, *sq2 = st + 288, *scale2 = st + 320, *shift2 = st + 352;
  float *ns1  = st + 384, *nq1 = st + 416, *nsc1   = st + 448, *nsh1   = st + 480;
  float *ns2  = st + 512, *nq2 = st + 544, *nsc2   = st + 576, *nsh2   = st + 608;

  pad_init_kernel<<<1024, 256, 0, stream>>>(x, xcur);

  for (int i = 0; i < 4; ++i) {
    hipMemsetAsync(st, 0, 640 * sizeof(float), stream);

    const float* W1p = Wah + i * 66 * 64;
    const float* b1p = bah + i * 64;
    const float* W2p = Wa  + i * 64 * 32;
    const float* b2p = ba  + i * 32;

    edge_pass_kernel<1><<<8192, 256, 0, stream>>>(xcur, W1p, b1p, W2p, b2p,
        scale1, shift1, scale2, shift2, sum1, sq1, sum2, sq2, agg);
    finalize_stats_kernel<<<1, 64, 0, stream>>>(sum1, sq1, g_eh + i * 64, b_eh + i * 64,
        scale1, shift1, 64, 1048576.f);
    edge_pass_kernel<2><<<8192, 256, 0, stream>>>(xcur, W1p, b1p, W2p, b2p,
        scale1, shift1, scale2, shift2, sum1, sq1, sum2, sq2, agg);
    finalize_stats_kernel<<<1, 64, 0, stream>>>(sum2, sq2, g_e + i * 32, b_e + i * 32,
        scale2, shift2, 32, 1048576.f);
    edge_pass_kernel<3><<<8192, 256, 0, stream>>>(xcur, W1p, b1p, W2p, b2p,
        scale1, shift1, scale2, shift2, sum1, sq1, sum2, sq2, agg);

    // node MLP: concat(agg, x) -> 32, BN; -> 32, BN; -> tanh linear
    node_layer_kernel<<<64, 256, 0, stream>>>(agg, xcur, nullptr, nullptr,
        Wn0 + i * 64 * 32, bn0 + i * 32, 64, h1, ns1, nq1);
    finalize_stats_kernel<<<1, 64, 0, stream>>>(ns1, nq1, g_n + i * 64, b_n + i * 64,
        nsc1, nsh1, 32, 8192.f);
    node_layer_kernel<<<64, 256, 0, stream>>>(h1, nullptr, nsc1, nsh1,
        Wn + i * 32 * 32, bnn + i * 32, 32, h2, ns2, nq2);
    finalize_stats_kernel<<<1, 64, 0, stream>>>(ns2, nq2, g_n + i * 64 + 32, b_n + i * 64 + 32,
        nsc2, nsh2, 32, 8192.f);
    node_layer_kernel<<<64, 256, 0, stream>>>(h2, nullptr, nsc2, nsh2,
        Wu + i * 32 * 32, bu + i * 32, 32, xcur, nullptr, nullptr);
  }

  hipMemcpyAsync(d_out, xcur, 262144 * sizeof(float), hipMemcpyDeviceToDevice, stream);
}